// ChebLayer_38998303048417
// MI455X (gfx1250) — compile-verified
//
#include <hip/hip_runtime.h>
#include <math.h>

typedef __attribute__((ext_vector_type(2))) float v2f;
typedef __attribute__((ext_vector_type(8))) float v8f;

#define DIM 128
#define KDIM 384
#define ROWS 32            // rows per block (100000 % 32 == 0 -> no tail)
#define LDS_STRIDE 388     // 388*4 bytes: 16B-aligned rows, conflict-free (388 mod 64 = 4)

// ---------------- degree + normalization ----------------
__global__ void degree_kernel(const int* __restrict__ dst, float* __restrict__ deg, int e) {
    int i = blockIdx.x * blockDim.x + threadIdx.x;
    if (i < e) atomicAdd(&deg[dst[i]], 1.0f);
}

__global__ void dsqrt_kernel(float* __restrict__ deg, int n) {
    int i = blockIdx.x * blockDim.x + threadIdx.x;
    if (i < n) {
        float d = deg[i];
        d = d < 1.0f ? 1.0f : d;          // jnp.clip(deg, 1.0)
        deg[i] = 1.0f / sqrtf(d);         // deg^-0.5 (in-place)
    }
}

// ---------------- fused scaled scatter-add: agg[dst] += dsqrt[src]*X[src] ----------------
__global__ __launch_bounds__(256) void scatter_kernel(
    const float* __restrict__ X, const float* __restrict__ dsqrt,
    const int* __restrict__ src, const int* __restrict__ dst,
    float* __restrict__ agg, int e)
{
    int edge = blockIdx.x * 8 + (threadIdx.x >> 5);
    if (edge >= e) return;
    int lane = threadIdx.x & 31;
    int s = src[edge];
    int d = dst[edge];
    float sc = dsqrt[s];
    const float4 xv = *(const float4*)(X + (size_t)s * DIM + lane * 4);
    float* a = agg + (size_t)d * DIM + lane * 4;
    atomicAdd(a + 0, xv.x * sc);
    atomicAdd(a + 1, xv.y * sc);
    atomicAdd(a + 2, xv.z * sc);
    atomicAdd(a + 3, xv.w * sc);
}

// ---------------- in-place combine: agg = alpha*dsqrt[row]*agg + beta*Xadd ----------------
__global__ void combine_kernel(float* __restrict__ agg, const float* __restrict__ dsqrt,
                               const float* __restrict__ Xadd, float alpha, float beta, int n)
{
    size_t i = (size_t)blockIdx.x * blockDim.x + threadIdx.x;
    if (i >= (size_t)n * DIM) return;
    int row = (int)(i >> 7);
    agg[i] = alpha * dsqrt[row] * agg[i] + beta * Xadd[i];
}

// ---------------- WMMA f32 GEMM: H = [X0|X1|X2] @ W + bias, scaled by snorm ----------------
// One block = 32 rows x all 128 output cols. 8 waves; each wave: 1 col-tile x 2 row-tiles.
// A-panel staged into LDS with CDNA5 async global->LDS b128 copies (ASYNCcnt).
__global__ __launch_bounds__(256) void gemm_kernel(
    const float* __restrict__ X0, const float* __restrict__ X1, const float* __restrict__ X2,
    const float* __restrict__ W,  const float* __restrict__ bias, const float* __restrict__ snorm,
    float* __restrict__ H, int n_rows)
{
    __shared__ float ldsA[ROWS * LDS_STRIDE];
    const int r0  = blockIdx.x * ROWS;
    const int tid = threadIdx.x;

    // ---- async stage of the 32x384 A-panel: 32 rows x 96 float4 chunks = 3072 chunks ----
    {
        const unsigned ldsBase = (unsigned)(uintptr_t)(void*)ldsA;  // LDS byte offset of panel
        #pragma unroll
        for (int c = 0; c < 12; ++c) {                 // 3072 / 256 threads = 12 each
            int chunk = c * 256 + tid;
            int row   = chunk / 96;                    // 0..31
            int seg   = chunk - row * 96;              // 0..95
            int which = seg >> 5;                      // 0..2 selects X0/X1/X2
            int col4  = seg & 31;                      // float4 index within 128 cols
            int grow  = r0 + row;
            if (grow >= n_rows) grow = n_rows - 1;     // clamp (stores guarded later)
            const float* gsrc = (which == 0 ? X0 : (which == 1 ? X1 : X2))
                              + (size_t)grow * DIM + col4 * 4;
            unsigned ldsOff = ldsBase + (unsigned)((row * LDS_STRIDE + which * 128 + col4 * 4) * 4);
            asm volatile("global_load_async_to_lds_b128 %0, %1, off"
                         :: "v"(ldsOff), "v"(gsrc)
                         : "memory");
        }
        asm volatile("s_wait_asynccnt 0" ::: "memory");
    }
    __syncthreads();

    const int lane = tid & 31;
    const int half = lane >> 4;        // lane half selects K sub-pair
    const int l16  = lane & 15;
    const int c0   = (tid >> 5) * 16;  // wave id * 16 = output column tile

    v8f acc0 = {};
    v8f acc1 = {};
    const float* Wc = W + c0;
    #pragma unroll 4
    for (int k0 = 0; k0 < KDIM; k0 += 4) {
        const int ka = k0 + (half ? 2 : 0);
        v2f b;
        // B 4x16 f32: lane = N; VGPR0 = K0/K2 by half, VGPR1 = K1/K3
        b.x = Wc[(size_t)ka * DIM + l16];
        b.y = Wc[(size_t)(ka + 1) * DIM + l16];
        // A 16x4 f32: lanes 0-15 rows M with K0/K1; lanes 16-31 K2/K3
        v2f a0, a1;
        a0.x = ldsA[l16 * LDS_STRIDE + ka];
        a0.y = ldsA[l16 * LDS_STRIDE + ka + 1];
        a1.x = ldsA[(16 + l16) * LDS_STRIDE + ka];
        a1.y = ldsA[(16 + l16) * LDS_STRIDE + ka + 1];
        acc0 = __builtin_amdgcn_wmma_f32_16x16x4_f32(false, a0, false, b,
                                                     (short)0, acc0, false, false);
        acc1 = __builtin_amdgcn_wmma_f32_16x16x4_f32(false, a1, false, b,
                                                     (short)0, acc1, false, false);
    }

    // ---- epilogue: uniform full-tile fast path (no per-lane predication) ----
    const float bval = bias[c0 + l16];
    if (r0 + ROWS <= n_rows) {
        #pragma unroll
        for (int j = 0; j < 8; ++j) {
            int row0 = r0 + j + half * 8;
            int row1 = row0 + 16;
            H[(size_t)row0 * DIM + c0 + l16] = (acc0[j] + bval) * snorm[row0];
            H[(size_t)row1 * DIM + c0 + l16] = (acc1[j] + bval) * snorm[row1];
        }
    } else {
        #pragma unroll
        for (int j = 0; j < 8; ++j) {
            int row0 = r0 + j + half * 8;
            int row1 = row0 + 16;
            if (row0 < n_rows) H[(size_t)row0 * DIM + c0 + l16] = (acc0[j] + bval) * snorm[row0];
            if (row1 < n_rows) H[(size_t)row1 * DIM + c0 + l16] = (acc1[j] + bval) * snorm[row1];
        }
    }
}

// ---------------- BN column statistics (sum, sumsq) ----------------
__global__ __launch_bounds__(128) void stats_kernel(const float* __restrict__ H,
                                                    float* __restrict__ stats, int n)
{
    int col = threadIdx.x;             // 128 threads = 128 columns
    float s = 0.0f, s2 = 0.0f;
    for (int r = blockIdx.x; r < n; r += gridDim.x) {
        float v = H[(size_t)r * DIM + col];
        s  += v;
        s2 += v * v;
    }
    atomicAdd(&stats[col],        s);
    atomicAdd(&stats[DIM + col],  s2);
}

// ---------------- BN normalize + affine + ReLU ----------------
__global__ void bn_relu_kernel(const float* __restrict__ H, const float* __restrict__ stats,
                               const float* __restrict__ gamma, const float* __restrict__ beta,
                               float* __restrict__ out, int n)
{
    size_t i = (size_t)blockIdx.x * blockDim.x + threadIdx.x;
    if (i >= (size_t)n * DIM) return;
    int col = (int)(i & (DIM - 1));
    float invn = 1.0f / (float)n;
    float mean = stats[col] * invn;
    float var  = stats[DIM + col] * invn - mean * mean;
    float v = (H[i] - mean) * rsqrtf(var + 1e-5f) * gamma[col] + beta[col];
    out[i] = v > 0.0f ? v : 0.0f;
}

extern "C" void kernel_launch(void* const* d_in, const int* in_sizes, int n_in,
                              void* d_out, int out_size, void* d_ws, size_t ws_size,
                              hipStream_t stream)
{
    const float* feature = (const float*)d_in[0];
    const float* snorm   = (const float*)d_in[1];
    const int*   src     = (const int*)d_in[2];
    const int*   dst     = (const int*)d_in[3];
    const float* weight  = (const float*)d_in[4];
    const float* bias    = (const float*)d_in[5];
    const float* gamma   = (const float*)d_in[6];
    const float* beta    = (const float*)d_in[7];

    const int n = in_sizes[1];   // snorm_n has N elements
    const int e = in_sizes[2];

    // workspace layout
    char* ws = (char*)d_ws;
    size_t off = 0;
    float* dsqrt = (float*)(ws + off); off += ((size_t)n * 4 + 255) & ~(size_t)255;
    float* X1    = (float*)(ws + off); off += (size_t)n * DIM * 4;
    float* X2    = (float*)(ws + off); off += (size_t)n * DIM * 4;
    float* H     = (float*)(ws + off); off += (size_t)n * DIM * 4;
    float* stats = (float*)(ws + off);

    const int tpb = 256;
    const size_t nd = (size_t)n * DIM;

    // 1) degree -> D^-1/2
    hipMemsetAsync(dsqrt, 0, (size_t)n * 4, stream);
    degree_kernel<<<(e + tpb - 1) / tpb, tpb, 0, stream>>>(dst, dsqrt, e);
    dsqrt_kernel<<<(n + tpb - 1) / tpb, tpb, 0, stream>>>(dsqrt, n);

    // 2) X1 = -Ahat @ X0      (re_norm == 1.0 with LAMBDA_MAX == 2.0)
    hipMemsetAsync(X1, 0, nd * 4, stream);
    scatter_kernel<<<(e + 7) / 8, 256, 0, stream>>>(feature, dsqrt, src, dst, X1, e);
    combine_kernel<<<(unsigned)((nd + tpb - 1) / tpb), tpb, 0, stream>>>(X1, dsqrt, feature, -1.0f, 0.0f, n);

    // 3) X2 = -2 * Ahat @ X1 - X0
    hipMemsetAsync(X2, 0, nd * 4, stream);
    scatter_kernel<<<(e + 7) / 8, 256, 0, stream>>>(X1, dsqrt, src, dst, X2, e);
    combine_kernel<<<(unsigned)((nd + tpb - 1) / tpb), tpb, 0, stream>>>(X2, dsqrt, feature, -2.0f, -1.0f, n);

    // 4) H = ([X0|X1|X2] @ W + bias) * snorm   (f32 WMMA, async-LDS staged)
    gemm_kernel<<<(n + ROWS - 1) / ROWS, 256, 0, stream>>>(feature, X1, X2, weight, bias, snorm, H, n);

    // 5) BatchNorm stats + normalize + ReLU
    hipMemsetAsync(stats, 0, 2 * DIM * 4, stream);
    stats_kernel<<<1024, 128, 0, stream>>>(H, stats, n);
    bn_relu_kernel<<<(unsigned)((nd + tpb - 1) / tpb), tpb, 0, stream>>>(H, stats, gamma, beta, (float*)d_out, n);
}